// YOLOLoss_14310831030489
// MI455X (gfx1250) — compile-verified
//
#include <hip/hip_runtime.h>
#include <math.h>

// ---- problem constants (from reference) ----
#define YG    13            // grid
#define YA    5             // anchors
#define YC    36            // classes
#define YT    30            // targets per batch
#define YCH   (5 + YC)      // 41 channels
#define YGG   (YG * YG)     // 169
#define YCPI  (YA * YGG)    // 845 cells per image

typedef float v2f __attribute__((ext_vector_type(2)));
typedef float v8f __attribute__((ext_vector_type(8)));

// Wave32 full-wave sum. Uses V_WMMA_F32_16X16X4_F32 with B = ones:
// A[m][0] = v_m (lanes 0-15, VGPR0), A[m][2] = v_{m+16} (lanes 16-31, VGPR0),
// A[m][1] = A[m][3] = 0  =>  D[m][n] = v_m + v_{m+16} for all n.
// D layout: lane<16 holds rows 0..7 in its 8 VGPRs, lane>=16 holds rows 8..15,
// so sum(D vgprs) + shfl_xor(...,16) = full 32-lane sum on every lane.
// EXEC must be all ones at call site (guaranteed by construction below).
__device__ __forceinline__ float wave_sum32(float v) {
#if defined(__gfx1250__) && __has_builtin(__builtin_amdgcn_wmma_f32_16x16x4_f32)
    v2f a;  a[0] = v;    a[1] = 0.0f;
    v2f bm; bm[0] = 1.0f; bm[1] = 1.0f;
    v8f c = {0.f, 0.f, 0.f, 0.f, 0.f, 0.f, 0.f, 0.f};
    v8f d = __builtin_amdgcn_wmma_f32_16x16x4_f32(
        /*neg_a=*/false, a, /*neg_b=*/false, bm,
        /*c_mod=*/(short)0, c, /*reuse_a=*/false, /*reuse_b=*/false);
    float s = ((d[0] + d[1]) + (d[2] + d[3])) + ((d[4] + d[5]) + (d[6] + d[7]));
    s += __shfl_xor(s, 16, 32);
    return s;
#else
    #pragma unroll
    for (int off = 16; off > 0; off >>= 1) v += __shfl_xor(v, off, 32);
    return v;
#endif
}

// Block-wide sum of three accumulators; result valid in thread 0.
__device__ __forceinline__ void block_sum3(float& a, float& b, float& c) {
    __shared__ float s0[8], s1[8], s2[8];
    a = wave_sum32(a);
    b = wave_sum32(b);
    c = wave_sum32(c);
    const int lane = threadIdx.x & 31;
    const int w    = threadIdx.x >> 5;
    if (lane == 0) { s0[w] = a; s1[w] = b; s2[w] = c; }
    __syncthreads();
    if (threadIdx.x == 0) {
        float t0 = 0.f, t1 = 0.f, t2 = 0.f;
        const int nw = (blockDim.x + 31) >> 5;
        for (int i = 0; i < nw; ++i) { t0 += s0[i]; t1 += s1[i]; t2 += s2[i]; }
        a = t0; b = t1; c = t2;
    }
}

// Pass 1: every cell, "no object" baseline.
//   conf:  (sigmoid(p0) * 1 - 0)^2
//   class: logsumexp(p5..p40) - p5          (== -log_softmax[class 0])
// Streams the full 142 MB of predictions exactly once, coalesced per channel.
__global__ __launch_bounds__(256) void yolo_cells_kernel(
        const float* __restrict__ pred, float* __restrict__ part, int Ntot) {
    const int n = blockIdx.x * 256 + threadIdx.x;
    const bool active = (n < Ntot);
    const int nc   = active ? n : (Ntot - 1);        // clamp: keep EXEC uniform for WMMA
    const int b    = nc / YCPI;
    const int cell = nc - b * YCPI;
    const int a    = cell / YGG;
    const int sp   = cell - a * YGG;                 // gj*13 + gi
    const float* base = pred + (size_t)(b * (YA * YCH) + a * YCH) * YGG + sp;

    const float conf  = 1.0f / (1.0f + expf(-base[0]));
    float lconf = conf * conf;

    float xv[YC];
    float m = -3.402823466e+38f;
    #pragma unroll
    for (int k = 0; k < YC; ++k) {
        xv[k] = base[(size_t)(5 + k) * YGG];
        m = fmaxf(m, xv[k]);
    }
    float s = 0.f;
    #pragma unroll
    for (int k = 0; k < YC; ++k) s += expf(xv[k] - m);
    float lclass = (m + logf(s)) - xv[0];

    if (!active) { lconf = 0.f; lclass = 0.f; }
    float lcoord = 0.f;
    block_sum3(lcoord, lconf, lclass);
    if (threadIdx.x == 0) {
        part[3 * blockIdx.x + 0] = lcoord;
        part[3 * blockIdx.x + 1] = lconf;
        part[3 * blockIdx.x + 2] = lclass;
    }
}

// Pass 2: one thread per batch image; serial (race-free, deterministic) target
// resolution + delta vs. the baseline pass.
__global__ __launch_bounds__(256) void yolo_targets_kernel(
        const float* __restrict__ pred, const float* __restrict__ tgt,
        float* __restrict__ part, int Bn) {
    const int bidx = blockIdx.x * 256 + threadIdx.x;
    const bool active = (bidx < Bn);
    const int b = active ? bidx : (Bn - 1);

    const float aw[YA] = {1.08f, 3.42f, 6.63f, 9.42f, 16.62f};
    const float ah[YA] = {1.19f, 4.41f, 11.38f, 5.11f, 10.52f};

    int   key[YT], cls[YT], val[YT];
    float tx[YT], ty[YT], tw[YT], th[YT];
    const float* tb = tgt + (size_t)b * YT * 5;

    for (int t = 0; t < YT; ++t) {
        const float x = tb[t * 5 + 0], y = tb[t * 5 + 1];
        const float w = tb[t * 5 + 2], h = tb[t * 5 + 3];
        const float cf = tb[t * 5 + 4];
        val[t] = ((x + y + w + h + cf) != 0.0f) ? 1 : 0;
        const float gx = x * YG, gy = y * YG, gw = w * YG, gh = h * YG;
        int gi = (int)gx; gi = (gi < 0) ? 0 : ((gi > YG - 1) ? YG - 1 : gi);
        int gj = (int)gy; gj = (gj < 0) ? 0 : ((gj > YG - 1) ? YG - 1 : gj);
        int best = 0; float bi = -1.0f;               // strict '>' => first max, like jnp.argmax
        #pragma unroll
        for (int a2 = 0; a2 < YA; ++a2) {
            const float inter = fminf(gw, aw[a2]) * fminf(gh, ah[a2]);
            const float uni   = gw * gh + aw[a2] * ah[a2] - inter;
            const float iou   = inter / (uni + 1e-16f);
            if (iou > bi) { bi = iou; best = a2; }
        }
        key[t] = (best * YG + gj) * YG + gi;          // cell index within image
        cls[t] = (int)cf;
        tx[t]  = gx - (float)gi;
        ty[t]  = gy - (float)gj;
        tw[t]  = logf(gw / aw[best] + 1e-16f);
        th[t]  = logf(gh / ah[best] + 1e-16f);
    }

    float lcoord = 0.f, lconf = 0.f, lclass = 0.f;
    for (int t = 0; t < YT; ++t) {
        if (!val[t]) continue;
        // last duplicate wins coords/conf; skip if a later target hits same cell
        bool later = false;
        for (int t2 = t + 1; t2 < YT; ++t2)
            later = later || (val[t2] && key[t2] == key[t]);
        if (later) continue;
        // duplicate tclass column-sets => argmax picks the smallest set class
        int minc = cls[t];
        for (int t2 = 0; t2 < YT; ++t2)
            if (val[t2] && key[t2] == key[t] && cls[t2] < minc) minc = cls[t2];

        const int k  = key[t];
        const int a2 = k / YGG;
        const int sp = k - a2 * YGG;
        const float* base = pred + (size_t)(b * (YA * YCH) + a2 * YCH) * YGG + sp;

        const float conf = 1.0f / (1.0f + expf(-base[0]));
        lconf += 25.0f * (conf - 1.f) * (conf - 1.f) - conf * conf;   // obj - baseline

        const float d1 = base[(size_t)1 * YGG] - tx[t];
        const float d2 = base[(size_t)2 * YGG] - ty[t];
        const float d3 = base[(size_t)3 * YGG] - tw[t];
        const float d4 = base[(size_t)4 * YGG] - th[t];
        lcoord += d1 * d1 + d2 * d2 + d3 * d3 + d4 * d4;

        // (-logp[minc]) - (-logp[0]) = x0 - x_minc  (LSE cancels)
        lclass += base[(size_t)5 * YGG] - base[(size_t)(5 + minc) * YGG];
    }

    if (!active) { lcoord = 0.f; lconf = 0.f; lclass = 0.f; }
    block_sum3(lcoord, lconf, lclass);
    if (threadIdx.x == 0) {
        part[3 * blockIdx.x + 0] = lcoord;
        part[3 * blockIdx.x + 1] = lconf;
        part[3 * blockIdx.x + 2] = lclass;
    }
}

// Final: deterministic fixed-order reduction of all block partials, scale by 1/B.
__global__ __launch_bounds__(256) void yolo_final_kernel(
        const float* __restrict__ part, int nPart, float invB,
        float* __restrict__ out) {
    float c = 0.f, f = 0.f, k = 0.f;
    for (int i = threadIdx.x; i < nPart; i += 256) {
        c += part[3 * i + 0];
        f += part[3 * i + 1];
        k += part[3 * i + 2];
    }
    block_sum3(c, f, k);
    if (threadIdx.x == 0) {
        c *= invB; f *= invB; k *= invB;
        out[0] = c + f + k;   // total_loss
        out[1] = c;           // loss_coord
        out[2] = f;           // loss_conf
        out[3] = k;           // loss_class
    }
}

extern "C" void kernel_launch(void* const* d_in, const int* in_sizes, int n_in,
                              void* d_out, int out_size, void* d_ws, size_t ws_size,
                              hipStream_t stream) {
    const float* pred = (const float*)d_in[0];
    const float* tgt  = (const float*)d_in[1];
    float* out  = (float*)d_out;
    float* part = (float*)d_ws;

    const int Bn   = in_sizes[0] / (YA * YCH * YGG);   // 1024
    const int Ntot = Bn * YCPI;                        // 865280
    const int nA   = (Ntot + 255) / 256;               // 3380 blocks
    const int nB   = (Bn + 255) / 256;                 // 4 blocks

    // Workspace layout: nA blocks * 3 floats, then nB blocks * 3 floats (~40 KB).
    float* partA = part;
    float* partB = part + 3 * (size_t)nA;

    yolo_cells_kernel<<<nA, 256, 0, stream>>>(pred, partA, Ntot);
    yolo_targets_kernel<<<nB, 256, 0, stream>>>(pred, tgt, partB, Bn);
    yolo_final_kernel<<<1, 256, 0, stream>>>(part, nA + nB, 1.0f / (float)Bn, out);
}